// Symptom2DiseaseGNN_5763846111927
// MI455X (gfx1250) — compile-verified
//
#include <hip/hip_runtime.h>
#include <hip/hip_bf16.h>

typedef __attribute__((ext_vector_type(2))) float v2f;
typedef __attribute__((ext_vector_type(8))) float v8f;

#define N_SYM 50000
#define N_DIS 10000
#define F_IN  128
#define HID   128
#define OUT_F 64
#define E_CNT 640000
#define NEG_SLOPE 0.2f

// ---------------- workspace layout (floats) ----------------
// acc  [N_SYM*128]      @ 0
// den  [N_SYM]          @ 6,400,000
// mkey [N_SYM] (uint)   @ 6,450,000
// hsrc [N_DIS*128]      @ 6,500,000
// al   [N_DIS]          @ 7,780,000
// ar   [N_SYM]          @ 7,790,000
// usrc [128]            @ 7,840,000
// udst [128]            @ 7,840,128
#define ACC_OFF   0
#define DEN_OFF   6400000
#define MKEY_OFF  6450000
#define HSRC_OFF  6500000
#define AL_OFF    7780000
#define AR_OFF    7790000
#define USRC_OFF  7840000
#define UDST_OFF  7840128
#define ZERO_F4   1625000   // (acc+den+mkey) = 6,500,000 floats / 4

// ---------------- zero init (acc, den, mkey) ----------------
__global__ void k_zero(float4* p) {
    int i = blockIdx.x * 256 + threadIdx.x;
    if (i < ZERO_F4) p[i] = make_float4(0.f, 0.f, 0.f, 0.f);
}

// ---------------- u = W @ a  (two 128x128 matvecs) ----------------
__global__ void k_make_u(const float* __restrict__ Ws, const float* __restrict__ as_,
                         const float* __restrict__ Wd, const float* __restrict__ ad,
                         float* __restrict__ usrc, float* __restrict__ udst) {
    int t = threadIdx.x;                 // 256 threads
    const float* W = (t < 128) ? Ws : Wd;
    const float* a = (t < 128) ? as_ : ad;
    float*       o = (t < 128) ? usrc : udst;
    int f = t & 127;
    float s = 0.f;
    #pragma unroll 4
    for (int h = 0; h < 128; ++h) s += W[f * 128 + h] * a[h];
    o[f] = s;
}

// ---------------- row . u  (al / ar), wave per row ----------------
__global__ void k_matvec128(const float* __restrict__ x, const float* __restrict__ u,
                            float* __restrict__ o, int nrows) {
    int wave = threadIdx.x >> 5, lane = threadIdx.x & 31;
    int row = blockIdx.x * 8 + wave;
    if (row >= nrows) return;
    float4 xv = ((const float4*)(x + (long)row * 128))[lane];
    float4 uv = ((const float4*)u)[lane];
    float s = xv.x * uv.x + xv.y * uv.y + xv.z * uv.z + xv.w * uv.w;
    #pragma unroll
    for (int off = 16; off; off >>= 1) s += __shfl_down(s, off, 32);
    if (lane == 0) o[row] = s;
}

// ---------------- h_src = x_dis @ Wsrc_ds  (WMMA f32 16x16x4) ----------------
__global__ void k_hsrc_gemm(const float* __restrict__ x, const float* __restrict__ W,
                            float* __restrict__ h) {
    __shared__ float sW[128 * 128];
    for (int i = threadIdx.x; i < 128 * 128; i += 256) sW[i] = W[i];
    __syncthreads();
    int wave = threadIdx.x >> 5, lane = threadIdx.x & 31;
    int tile = blockIdx.x * 8 + wave;
    if (tile >= N_DIS / 16) return;
    int m  = lane & 15;
    int kh = (lane >> 4) * 2;            // K offset: lanes 0-15 -> K0/K1, 16-31 -> K2/K3
    int row = tile * 16 + m;
    v8f acc[8];
    #pragma unroll
    for (int i = 0; i < 8; ++i) acc[i] = (v8f){0.f,0.f,0.f,0.f,0.f,0.f,0.f,0.f};
    const float* xr = x + (long)row * 128;
    for (int k = 0; k < 128; k += 4) {
        v2f a; a.x = xr[k + kh]; a.y = xr[k + kh + 1];
        #pragma unroll
        for (int nt = 0; nt < 8; ++nt) {
            int col = m + nt * 16;
            v2f b; b.x = sW[(k + kh) * 128 + col]; b.y = sW[(k + kh + 1) * 128 + col];
            acc[nt] = __builtin_amdgcn_wmma_f32_16x16x4_f32(
                false, a, false, b, (short)0, acc[nt], false, false);
        }
    }
    int rowbase = tile * 16 + ((lane >> 4) << 3);   // lanes>=16 hold M=8..15
    #pragma unroll
    for (int nt = 0; nt < 8; ++nt) {
        int col = m + nt * 16;
        #pragma unroll
        for (int r = 0; r < 8; ++r)
            h[(long)(rowbase + r) * 128 + col] = acc[nt][r];
    }
}

// ---------------- edge pass 1: per-dst max (ordered-uint atomicMax) ----------------
__device__ __forceinline__ unsigned f2key(float f) {
    unsigned u = __float_as_uint(f);
    return (u & 0x80000000u) ? ~u : (u | 0x80000000u);
}
__device__ __forceinline__ float key2f(unsigned k) {
    return (k & 0x80000000u) ? __uint_as_float(k & 0x7fffffffu) : __uint_as_float(~k);
}

__global__ void k_edge_max(const int* __restrict__ esrc, const int* __restrict__ edst,
                           const float* __restrict__ al, const float* __restrict__ ar,
                           unsigned* __restrict__ mkey) {
    int i = blockIdx.x * 256 + threadIdx.x;
    if (i >= E_CNT) return;
    int d = edst[i];                     // disease (source side of ds-GAT)
    int s = esrc[i];                     // symptom (destination side)
    float e = al[d] + ar[s];
    e = e > 0.f ? e : NEG_SLOPE * e;
    atomicMax(&mkey[s], f2key(e));
}

// ---------------- edge pass 2: exp + weighted scatter-accumulate ----------------
__global__ void k_edge_acc(const int* __restrict__ esrc, const int* __restrict__ edst,
                           const float* __restrict__ al, const float* __restrict__ ar,
                           const unsigned* __restrict__ mkey, const float* __restrict__ hsrc,
                           float* __restrict__ acc, float* __restrict__ den) {
    int wave = threadIdx.x >> 5, lane = threadIdx.x & 31;
    int eid = blockIdx.x * 8 + wave;     // one edge per wave
    if (eid >= E_CNT) return;
    int d = edst[eid];
    int s = esrc[eid];
    float e = al[d] + ar[s];
    e = e > 0.f ? e : NEG_SLOPE * e;
    float ex = __expf(e - key2f(mkey[s]));
    if (lane == 0) atomicAdd(&den[s], ex);
    float4 h = ((const float4*)(hsrc + (long)d * 128))[lane];
    float* ap = acc + (long)s * 128 + lane * 4;
    atomicAdd(ap + 0, ex * h.x);
    atomicAdd(ap + 1, ex * h.y);
    atomicAdd(ap + 2, ex * h.z);
    atomicAdd(ap + 3, ex * h.w);
}

// ---------------- out = relu(acc/den + b_ds) @ lin_W + lin_b  (WMMA f32) ----------------
__global__ void k_final_gemm(const float* __restrict__ acc, const float* __restrict__ den,
                             const float* __restrict__ b_ds, const float* __restrict__ linW,
                             const float* __restrict__ linb, float* __restrict__ out) {
    __shared__ float sW[128 * 64];
    for (int i = threadIdx.x; i < 128 * 64; i += 256) sW[i] = linW[i];
    __syncthreads();
    int wave = threadIdx.x >> 5, lane = threadIdx.x & 31;
    int tile = blockIdx.x * 8 + wave;
    if (tile >= N_SYM / 16) return;
    int m  = lane & 15;
    int kh = (lane >> 4) * 2;
    int row = tile * 16 + m;
    float dd = den[row];
    float rden = dd > 0.f ? 1.f / dd : 0.f;   // empty segment: acc==0 -> h = b_ds (matches ref)
    v8f c[4];
    #pragma unroll
    for (int i = 0; i < 4; ++i) c[i] = (v8f){0.f,0.f,0.f,0.f,0.f,0.f,0.f,0.f};
    const float* arow = acc + (long)row * 128;
    for (int k = 0; k < 128; k += 4) {
        float a0 = fmaf(arow[k + kh],     rden, b_ds[k + kh]);
        float a1 = fmaf(arow[k + kh + 1], rden, b_ds[k + kh + 1]);
        v2f a; a.x = a0 > 0.f ? a0 : 0.f; a.y = a1 > 0.f ? a1 : 0.f;   // ReLU prologue
        #pragma unroll
        for (int nt = 0; nt < 4; ++nt) {
            int col = m + nt * 16;
            v2f b; b.x = sW[(k + kh) * 64 + col]; b.y = sW[(k + kh + 1) * 64 + col];
            c[nt] = __builtin_amdgcn_wmma_f32_16x16x4_f32(
                false, a, false, b, (short)0, c[nt], false, false);
        }
    }
    int rowbase = tile * 16 + ((lane >> 4) << 3);
    #pragma unroll
    for (int nt = 0; nt < 4; ++nt) {
        int col = m + nt * 16;
        float lb = linb[col];
        #pragma unroll
        for (int r = 0; r < 8; ++r)
            out[(long)(rowbase + r) * 64 + col] = c[nt][r] + lb;
    }
}

extern "C" void kernel_launch(void* const* d_in, const int* in_sizes, int n_in,
                              void* d_out, int out_size, void* d_ws, size_t ws_size,
                              hipStream_t stream) {
    const float* x_sym   = (const float*)d_in[0];
    const float* x_dis   = (const float*)d_in[1];
    const int*   esrc    = (const int*)d_in[2];
    const int*   edst    = (const int*)d_in[3];
    // d_in[4..8] = sd-branch params: dead code (h_dis unused by the reference return)
    const float* Wsrc_ds = (const float*)d_in[9];
    const float* Wdst_ds = (const float*)d_in[10];
    const float* asrc_ds = (const float*)d_in[11];
    const float* adst_ds = (const float*)d_in[12];
    const float* b_ds    = (const float*)d_in[13];
    const float* lin_W   = (const float*)d_in[14];
    const float* lin_b   = (const float*)d_in[15];
    float* outp = (float*)d_out;

    float* ws = (float*)d_ws;
    float*    acc  = ws + ACC_OFF;
    float*    den  = ws + DEN_OFF;
    unsigned* mkey = (unsigned*)(ws + MKEY_OFF);
    float*    hsrc = ws + HSRC_OFF;
    float*    al   = ws + AL_OFF;
    float*    ar   = ws + AR_OFF;
    float*    usrc = ws + USRC_OFF;
    float*    udst = ws + UDST_OFF;

    // 1) zero acc/den/mkey
    k_zero<<<(ZERO_F4 + 255) / 256, 256, 0, stream>>>((float4*)ws);
    // 2) u vectors (fold h_dst GEMM and al projection into matvecs)
    k_make_u<<<1, 256, 0, stream>>>(Wsrc_ds, asrc_ds, Wdst_ds, adst_ds, usrc, udst);
    // 3) h_src = x_dis @ Wsrc_ds   (WMMA)
    k_hsrc_gemm<<<(N_DIS / 16 + 7) / 8, 256, 0, stream>>>(x_dis, Wsrc_ds, hsrc);
    // 4) al = x_dis @ usrc ; ar = x_sym @ udst
    k_matvec128<<<(N_DIS + 7) / 8, 256, 0, stream>>>(x_dis, usrc, al, N_DIS);
    k_matvec128<<<(N_SYM + 7) / 8, 256, 0, stream>>>(x_sym, udst, ar, N_SYM);
    // 5) segment max over symptom nodes
    k_edge_max<<<(E_CNT + 255) / 256, 256, 0, stream>>>(esrc, edst, al, ar, mkey);
    // 6) exp + accumulate numerator/denominator
    k_edge_acc<<<E_CNT / 8, 256, 0, stream>>>(esrc, edst, al, ar, mkey, hsrc, acc, den);
    // 7) normalize + bias + ReLU + final linear  (WMMA)
    k_final_gemm<<<(N_SYM / 16 + 7) / 8, 256, 0, stream>>>(acc, den, b_ds, lin_W, lin_b, outp);
}